// Decoder_29033978921192
// MI455X (gfx1250) — compile-verified
//
#include <hip/hip_runtime.h>
#include <hip/hip_bf16.h>

// ---------------------------------------------------------------------------
// Decoder step (local-attention LSTM) for MI455X / gfx1250, fp32 via
// V_WMMA_F32_16X16X4_F32.  Memory bound: fc2_w (205 MB) dominates.
// Weights stream HBM -> LDS via GLOBAL_LOAD_ASYNC_TO_LDS_B128 (ASYNCcnt
// DMA) once per column tile; A fragments are register-resident per K chunk
// so the inner loop is pure ds_load + wmma.
// ---------------------------------------------------------------------------

typedef __attribute__((ext_vector_type(2))) float v2f;
typedef __attribute__((ext_vector_type(8))) float v8f;
typedef int v4i_gcc __attribute__((__vector_size__(16)));   // matches builtin param

#define HDIM   1024
#define BDIM   64
#define VDIM   50000
#define WWIN   10
#define WLEN   21
#define SENC   1021
#define HHDIM  512
#define LMAX   1000

#define AS1 __attribute__((address_space(1)))
#define AS3 __attribute__((address_space(3)))

#if defined(__has_builtin)
# if __has_builtin(__builtin_amdgcn_global_load_async_to_lds_b128)
#  define HAVE_ASYNC_TO_LDS 1
# endif
# if __has_builtin(__builtin_amdgcn_s_wait_asynccnt)
#  define HAVE_WAIT_ASYNC 1
# endif
#endif

__device__ __forceinline__ void async_wait0() {
#if defined(HAVE_WAIT_ASYNC)
    __builtin_amdgcn_s_wait_asynccnt(0);
#else
    asm volatile("s_wait_asynccnt 0" ::: "memory");
#endif
}

// async 16B global -> LDS copy (per lane), tracked by ASYNCcnt
__device__ __forceinline__ void async_copy_b128(const float* g, float* l) {
#if defined(HAVE_ASYNC_TO_LDS)
    __builtin_amdgcn_global_load_async_to_lds_b128(
        (AS1 v4i_gcc*)g, (AS3 v4i_gcc*)l, 0, 0);
#else
    const float4 w = *(const float4*)g;
    l[0] = w.x; l[1] = w.y; l[2] = w.z; l[3] = w.w;
#endif
}

// ---------------------------------------------------------------------------
// Generic WMMA GEMM:  Y[64, N] = act( A1 @ W1^T (+ A2 @ W2^T) + b1 (+ b2) )
//   - W* stored row-major (N, K)  (already B^T for the WMMA B operand)
//   - block = 128 threads = 4 waves; block owns one 16-wide N tile,
//     wave w owns M tile [16w, 16w+16)
//   - weight tile chunks staged HBM->LDS (async b128),
//     A fragments batch-loaded into registers per chunk (L2-resident)
// ---------------------------------------------------------------------------
template <bool GATHER, bool DUAL, bool TANH_ACT>
__global__ __launch_bounds__(128) void wmma_gemm(
    const float* __restrict__ A1, int lda1, const int* __restrict__ gidx,
    const float* __restrict__ W1, int ldw1,
    const float* __restrict__ A2, int lda2,
    const float* __restrict__ W2, int ldw2,
    const float* __restrict__ b1, const float* __restrict__ b2,
    float* __restrict__ Y, int ldy, int K)
{
    constexpr int KC  = 64;         // K chunk staged in LDS
    constexpr int KCP = KC + 4;     // pad: keeps rows 16B aligned, banks spread
    constexpr int NSTEP = KC / 4;   // wmma k-steps per chunk
    __shared__ float w1s[16 * KCP];
    __shared__ float w2s[DUAL ? 16 * KCP : 1];

    const int tid     = threadIdx.x;
    const int lane    = tid & 31;
    const int wave    = tid >> 5;       // m-tile
    const int n0      = blockIdx.x * 16;
    const int m0      = wave * 16;
    const int halfsel = lane >> 4;      // 0 for lanes 0-15, 1 for 16-31
    const int l16     = lane & 15;

    // per-lane A row pointers (A fragment: row = m0 + lane%16)
    const int arow = m0 + l16;
    const float* a1row;
    if (GATHER) a1row = A1 + (size_t)gidx[arow] * lda1;
    else        a1row = A1 + (size_t)arow * lda1;
    const float* a2row = nullptr;
    if (DUAL)   a2row = A2 + (size_t)arow * lda2;

    // staging geometry: 16 rows x KC floats = 256 float4 slots, 2 per thread
    const int s0r = tid >> 4;              // slot row for i=0  (tid/16)
    const int s0c = (tid & 15) * 4;        // slot col (floats)

    v8f acc = {};

    for (int kc = 0; kc < K; kc += KC) {
        // ---- stage W chunk(s) into LDS (async DMA, ASYNCcnt tracked) ----
        #pragma unroll
        for (int i = 0; i < 2; ++i) {
            const int r = s0r + i * 8;     // +128 threads -> +8 rows
            async_copy_b128(W1 + (size_t)(n0 + r) * ldw1 + kc + s0c,
                            &w1s[r * KCP + s0c]);
            if constexpr (DUAL) {
                async_copy_b128(W2 + (size_t)(n0 + r) * ldw2 + kc + s0c,
                                &w2s[r * KCP + s0c]);
            }
        }

        // ---- batch-load A fragments for the whole chunk into registers ----
        v2f a1f[NSTEP];
        v2f a2f[DUAL ? NSTEP : 1];
        #pragma unroll
        for (int s = 0; s < NSTEP; ++s) {
            a1f[s] = *(const v2f*)(a1row + kc + 4 * s + 2 * halfsel);
            if constexpr (DUAL)
                a2f[s] = *(const v2f*)(a2row + kc + 4 * s + 2 * halfsel);
        }

#if defined(HAVE_ASYNC_TO_LDS)
        async_wait0();
#endif
        __syncthreads();

        // ---- inner loop: pure LDS reads + WMMA ----
        #pragma unroll
        for (int s = 0; s < NSTEP; ++s) {
            const int kk = 4 * s + 2 * halfsel;
            v2f bv1 = *(const v2f*)(&w1s[l16 * KCP + kk]);
            acc = __builtin_amdgcn_wmma_f32_16x16x4_f32(
                false, a1f[s], false, bv1, (short)0, acc, false, false);
            if constexpr (DUAL) {
                v2f bv2 = *(const v2f*)(&w2s[l16 * KCP + kk]);
                acc = __builtin_amdgcn_wmma_f32_16x16x4_f32(
                    false, a2f[s], false, bv2, (short)0, acc, false, false);
            }
        }
        __syncthreads();
    }

    // epilogue: C/D layout -> row = m0 + v + 8*halfsel, col = n0 + lane%16
    const int coln = n0 + l16;
    float bias = b1[coln];
    if (b2) bias += b2[coln];
    #pragma unroll
    for (int v = 0; v < 8; ++v) {
        float val = acc[v] + bias;
        if (TANH_ACT) val = tanhf(val);
        const int rowm = m0 + v + 8 * halfsel;
        Y[(size_t)rowm * ldy + coln] = val;
    }
}

// ---------------------------------------------------------------------------
// LSTM gate nonlinearities: g (B,4H) gate order i,f,g,o -> h,c (B,H)
// ---------------------------------------------------------------------------
__global__ __launch_bounds__(256) void lstm_gates(
    const float* __restrict__ g, const float* __restrict__ c_prev,
    float* __restrict__ h_out, float* __restrict__ c_out)
{
    const int t = blockIdx.x * blockDim.x + threadIdx.x;
    if (t >= BDIM * HDIM) return;
    const int b = t / HDIM, j = t % HDIM;
    const float* gb = g + (size_t)b * 4 * HDIM;
    const float zi = gb[j];
    const float zf = gb[HDIM + j];
    const float zg = gb[2 * HDIM + j];
    const float zo = gb[3 * HDIM + j];
    const float i = 1.f / (1.f + expf(-zi));
    const float f = 1.f / (1.f + expf(-zf));
    const float o = 1.f / (1.f + expf(-zo));
    const float c = f * c_prev[t] + i * tanhf(zg);
    h_out[t] = o * tanhf(c);
    c_out[t] = c;
}

// ---------------------------------------------------------------------------
// Attention: window center p, scores over 21 positions, softmax*gaussian,
// context.  One block (256 thr) per batch element.
// ---------------------------------------------------------------------------
__global__ __launch_bounds__(256) void attn_kernel(
    const float* __restrict__ t1,        // (B,HH) = tanh(h_t @ afc1^T + b)
    const float* __restrict__ afc2_w, const float* __restrict__ afc2_b,
    const float* __restrict__ h_t,       // (B,H)
    const float* __restrict__ enc,       // (S_ENC,B,H)
    const int* __restrict__ lengths,
    float* __restrict__ a_out,           // (B,WLEN)
    int* __restrict__ start_out,         // (B,)
    float* __restrict__ ctx)             // (B,H)
{
    __shared__ float red[256];
    __shared__ float sc[WLEN];
    __shared__ float sa[WLEN];
    __shared__ float s_p;
    __shared__ int   s_start;

    const int b = blockIdx.x, t = threadIdx.x;

    // p = W + len * sigmoid(t1[b] . afc2_w + afc2_b)
    float part = 0.f;
    for (int j = t; j < HHDIM; j += 256) part += t1[(size_t)b * HHDIM + j] * afc2_w[j];
    red[t] = part; __syncthreads();
    for (int s = 128; s > 0; s >>= 1) { if (t < s) red[t] += red[t + s]; __syncthreads(); }
    if (t == 0) {
        const float z  = red[0] + afc2_b[0];
        const float pb = 1.f / (1.f + expf(-z));
        const float lf = (float)lengths[b];
        const float p  = (float)WWIN + lf * pb;
        s_p = p;
        s_start = (int)rintf(p - (float)WWIN);   // round-half-even like jnp.round
    }
    __syncthreads();
    const float p     = s_p;
    const int   start = s_start;
    const float lf    = (float)lengths[b];

    // scores: h_t[b] . enc[start+w, b, :]
    for (int w = 0; w < WLEN; ++w) {
        const int sidx = start + w;
        const float* row = enc + ((size_t)sidx * BDIM + b) * HDIM;
        float ps = 0.f;
        for (int j = t; j < HDIM; j += 256) ps += h_t[(size_t)b * HDIM + j] * row[j];
        red[t] = ps; __syncthreads();
        for (int s = 128; s > 0; s >>= 1) { if (t < s) red[t] += red[t + s]; __syncthreads(); }
        if (t == 0) {
            const float pos = (float)sidx;
            const bool mask = (pos < (float)WWIN) || (pos >= lf + (float)WWIN);
            sc[w] = mask ? 1e-14f : red[0];
        }
        __syncthreads();
    }

    if (t == 0) {
        float mx = sc[0];
        for (int w = 1; w < WLEN; ++w) mx = fmaxf(mx, sc[w]);
        float e[WLEN], sum = 0.f;
        for (int w = 0; w < WLEN; ++w) { e[w] = expf(sc[w] - mx); sum += e[w]; }
        for (int w = 0; w < WLEN; ++w) {
            const float d = (float)(start + w) - p;
            const float gauss = expf(-d * d / 50.f);   // 2*(W/2)^2 = 50
            sa[w] = (e[w] / sum) * gauss;
        }
    }
    __syncthreads();
    if (t < WLEN) a_out[b * WLEN + t] = sa[t];
    if (t == 0)   start_out[b] = start;

    // ctx[b, :] = sum_w a[w] * enc[start+w, b, :]
    for (int j = t; j < HDIM; j += 256) {
        float accv = 0.f;
        #pragma unroll
        for (int w = 0; w < WLEN; ++w) {
            accv += sa[w] * enc[((size_t)(start + w) * BDIM + b) * HDIM + j];
        }
        ctx[(size_t)b * HDIM + j] = accv;
    }
}

// ---------------------------------------------------------------------------
// Scatter sample-0 attention into the (1, s0=1000) weights buffer.
// ---------------------------------------------------------------------------
__global__ __launch_bounds__(1024) void weights_kernel(
    const float* __restrict__ a0, const int* __restrict__ start, float* __restrict__ wts)
{
    const int t = threadIdx.x;
    if (t < LMAX) wts[t] = 0.f;
    __syncthreads();
    if (t < WLEN) {
        const int idx0 = start[0] + t - WWIN;
        if (idx0 >= 0 && idx0 < LMAX) wts[idx0] = a0[t];
    }
}

// ---------------------------------------------------------------------------
extern "C" void kernel_launch(void* const* d_in, const int* in_sizes, int n_in,
                              void* d_out, int out_size, void* d_ws, size_t ws_size,
                              hipStream_t stream) {
    (void)in_sizes; (void)n_in; (void)out_size; (void)ws_size;

    const float* enc     = (const float*)d_in[1];
    const int*   word    = (const int*)  d_in[2];
    const float* h0      = (const float*)d_in[3];
    const float* c0      = (const float*)d_in[4];
    const int*   lengths = (const int*)  d_in[6];
    const float* emb     = (const float*)d_in[7];
    const float* Wih0    = (const float*)d_in[8];
    const float* Whh0    = (const float*)d_in[9];
    const float* bih0    = (const float*)d_in[10];
    const float* bhh0    = (const float*)d_in[11];
    const float* Wih1    = (const float*)d_in[12];
    const float* Whh1    = (const float*)d_in[13];
    const float* bih1    = (const float*)d_in[14];
    const float* bhh1    = (const float*)d_in[15];
    const float* afc1_w  = (const float*)d_in[16];
    const float* afc1_b  = (const float*)d_in[17];
    const float* afc2_w  = (const float*)d_in[18];
    const float* afc2_b  = (const float*)d_in[19];
    const float* fc1_w   = (const float*)d_in[20];
    const float* fc1_b   = (const float*)d_in[21];
    const float* fc2_w   = (const float*)d_in[22];
    const float* fc2_b   = (const float*)d_in[23];

    float* outp = (float*)d_out;
    // output slices (flat, return order: y, out, h_n, c_n, weights)
    float* y_out   = outp;                                   // 64*50000
    float* out_fc1 = outp + (size_t)BDIM * VDIM;             // 1*64*1024
    float* h_n     = out_fc1 + (size_t)BDIM * HDIM;          // 2*64*1024
    float* c_n     = h_n + 2 * (size_t)BDIM * HDIM;          // 2*64*1024
    float* wts     = c_n + 2 * (size_t)BDIM * HDIM;          // 1000
    float* h_l0 = h_n;
    float* h_l1 = h_n + (size_t)BDIM * HDIM;                 // h_t
    float* c_l0 = c_n;
    float* c_l1 = c_n + (size_t)BDIM * HDIM;

    // workspace slices
    float* ws   = (float*)d_ws;
    float* g0   = ws;                                        // 64*4096
    float* g1   = g0 + (size_t)BDIM * 4 * HDIM;              // 64*4096
    float* t1   = g1 + (size_t)BDIM * 4 * HDIM;              // 64*512
    float* a_w  = t1 + (size_t)BDIM * HHDIM;                 // 64*21 (pad 32)
    float* ctx  = a_w + (size_t)BDIM * 32;                   // 64*1024
    int*   st   = (int*)(ctx + (size_t)BDIM * HDIM);         // 64 ints

    const dim3 blk128(128), blk256(256);

    // LSTM layer 0: g0 = emb[word] @ Wih0^T + h0[0] @ Whh0^T + biases
    wmma_gemm<true, true, false><<<4 * HDIM / 16, blk128, 0, stream>>>(
        emb, HDIM, word, Wih0, HDIM, h0, HDIM, Whh0, HDIM,
        bih0, bhh0, g0, 4 * HDIM, HDIM);
    lstm_gates<<<BDIM * HDIM / 256, blk256, 0, stream>>>(g0, c0, h_l0, c_l0);

    // LSTM layer 1: g1 = h_l0 @ Wih1^T + h0[1] @ Whh1^T + biases
    wmma_gemm<false, true, false><<<4 * HDIM / 16, blk128, 0, stream>>>(
        h_l0, HDIM, nullptr, Wih1, HDIM, h0 + (size_t)BDIM * HDIM, HDIM, Whh1, HDIM,
        bih1, bhh1, g1, 4 * HDIM, HDIM);
    lstm_gates<<<BDIM * HDIM / 256, blk256, 0, stream>>>(
        g1, c0 + (size_t)BDIM * HDIM, h_l1, c_l1);

    // attention fc1: t1 = tanh(h_t @ afc1_w^T + afc1_b)
    wmma_gemm<false, false, true><<<HHDIM / 16, blk128, 0, stream>>>(
        h_l1, HDIM, nullptr, afc1_w, HDIM, nullptr, 0, nullptr, 0,
        afc1_b, nullptr, t1, HHDIM, HDIM);

    // attention window + softmax*gaussian + context
    attn_kernel<<<BDIM, blk256, 0, stream>>>(
        t1, afc2_w, afc2_b, h_l1, enc, lengths, a_w, st, ctx);

    // sample-0 weights scatter (a is stored with row stride WLEN)
    weights_kernel<<<1, 1024, 0, stream>>>(a_w, st, wts);

    // fc1: out = tanh([ctx, h_t] @ fc1_w^T + fc1_b)  (split fc1_w columns)
    wmma_gemm<false, true, true><<<HDIM / 16, blk128, 0, stream>>>(
        ctx, HDIM, nullptr, fc1_w, 2 * HDIM, h_l1, HDIM, fc1_w + HDIM, 2 * HDIM,
        fc1_b, nullptr, out_fc1, HDIM, HDIM);

    // fc2: y = out @ fc2_w^T + fc2_b   (the 205 MB stream; 3125 column tiles)
    wmma_gemm<false, false, false><<<VDIM / 16, blk128, 0, stream>>>(
        out_fc1, HDIM, nullptr, fc2_w, HDIM, nullptr, 0, nullptr, 0,
        fc2_b, nullptr, y_out, VDIM, HDIM);
}